// EdgeConv_33998961115888
// MI455X (gfx1250) — compile-verified
//
#include <hip/hip_runtime.h>
#include <hip/hip_bf16.h>

#define Bb 8
#define Cc 64
#define Nn 4096
#define Oo 64
#define Kk 20

typedef float v2f __attribute__((ext_vector_type(2)));
typedef float v8f __attribute__((ext_vector_type(8)));

// ---------------------------------------------------------------------------
// CDNA5 f32 WMMA: D(16x16) = A(16x4) * B(4x16) + C   (V_WMMA_F32_16X16X4_F32)
// ---------------------------------------------------------------------------
__device__ __forceinline__ v8f wmma4(v2f a, v2f b, v8f c) {
  return __builtin_amdgcn_wmma_f32_16x16x4_f32(
      /*neg_a=*/false, a, /*neg_b=*/false, b,
      /*c_mod=*/(short)0, c, /*reuse_a=*/false, /*reuse_b=*/false);
}

// Fully-unrolled register-resident sorted insert (ascending; vals[0] = min).
__device__ __forceinline__ void topk_insert(float (&vals)[Kk], int (&ids)[Kk],
                                            float v, int ci) {
  if (v <= vals[0]) return;
  float cv = v; int cx = ci;
#pragma unroll
  for (int i = Kk - 1; i >= 0; --i) {
    bool m = cv > vals[i];
    float tv = vals[i]; int ti = ids[i];
    vals[i] = m ? cv : tv;
    ids[i]  = m ? cx : ti;
    cv = m ? tv : cv;
    cx = m ? ti : cx;
  }
}

// ---------------------------------------------------------------------------
// K0a: transpose x (B,C,N) -> xt (B,N,C) and compute row norms xx[b,n]
// ---------------------------------------------------------------------------
__global__ __launch_bounds__(256) void prep_x_kernel(const float* __restrict__ x,
                                                     float* __restrict__ xt,
                                                     float* __restrict__ xx) {
  const int p = blockIdx.x * 256 + threadIdx.x;   // 0 .. B*N-1
  const int b = p >> 12;
  const int n = p & (Nn - 1);
  float ss = 0.f;
  float* dst = xt + (size_t)p * Cc;
  for (int c = 0; c < Cc; ++c) {
    float v = x[((size_t)(b * Cc + c)) * Nn + n];
    ss += v * v;
    dst[c] = v;
  }
  xx[p] = ss;
}

// ---------------------------------------------------------------------------
// K0b: Wcomb[o][c] = c<64 ? W1[o][c] : (W2-W1)[o][c-64]; zero global stats
// ---------------------------------------------------------------------------
__global__ __launch_bounds__(256) void prep_w_kernel(const float* __restrict__ W,
                                                     float* __restrict__ Wt,
                                                     float* __restrict__ gstats) {
  const int t = threadIdx.x;
  for (int i = t; i < Oo * 128; i += 256) {
    int o = i >> 7, c = i & 127;
    float v = W[o * 128 + c];
    Wt[i] = (c < 64) ? v : (v - W[o * 128 + (c - 64)]);
  }
  if (t < 128) gstats[t] = 0.f;
}

// ---------------------------------------------------------------------------
// K1: fused pairwise neg-distance GEMM (f32 WMMA) + per-row top-20 selection.
// One workgroup = 32 query rows x all 4096 columns (64-col blocks).
// Candidate tiles are double-buffered in LDS via async global->LDS DMA
// (GLOBAL_LOAD_ASYNC_TO_LDS_B128, ASYNCcnt) overlapping with the WMMA work.
// ---------------------------------------------------------------------------
#define SA_STRIDE 68   // padded so 16-lane stride-row fragment loads are conflict-free
#define SB_STRIDE 68
#define SD_STRIDE 65

__global__ __launch_bounds__(128) void dist_topk_kernel(const float* __restrict__ xt,
                                                        const float* __restrict__ xx,
                                                        int* __restrict__ idx_out) {
  // One arena, manually carved; merge buffers alias the dead B buffers.
  __shared__ __align__(16) float smem[32 * SA_STRIDE + 2 * 64 * SB_STRIDE +
                                      32 * SD_STRIDE + 32];
  float* sA   = smem;                       // 32 x 68
  float* sB0  = sA + 32 * SA_STRIDE;        // 64 x 68 (async dst, buffer 0)
  float* sB1  = sB0 + 64 * SB_STRIDE;       // 64 x 68 (async dst, buffer 1)
  float* sD   = sB1 + 64 * SB_STRIDE;       // 32 x 65
  float* sxxA = sD + 32 * SD_STRIDE;        // 32
  float* mVal = sB0;                        // aliased: used only after main loop
  int*   mIdx = (int*)(sB0 + 32 * 4 * Kk);  // spills into sB1 region (also dead)

  const int t    = threadIdx.x;
  const int b    = blockIdx.y;
  const int row0 = blockIdx.x * 32;
  const int wave = t >> 5;
  const int lane = t & 31;
  const int lm   = lane & 15;
  const int lh   = lane >> 4;

  const float* xtb = xt + (size_t)b * Nn * Cc;   // batch base for async SADDR

  // Stage the 32 query rows (regular loads; done once).
  for (int f = t; f < 32 * 16; f += 128) {
    int m = f >> 4, c4 = (f & 15) << 2;
    const float4 v = *(const float4*)(xtb + (size_t)(row0 + m) * Cc + c4);
    sA[m * SA_STRIDE + c4 + 0] = v.x;
    sA[m * SA_STRIDE + c4 + 1] = v.y;
    sA[m * SA_STRIDE + c4 + 2] = v.z;
    sA[m * SA_STRIDE + c4 + 3] = v.w;
  }
  if (t < 32) sxxA[t] = xx[b * Nn + row0 + t];

  // Per-lane chunk mapping for the async tile copy: 64 rows x 64 floats
  // = 1024 x 16B chunks; each wave issues 8 async B128 ops (ASYNCcnt <= 8).
  const int cbase16 = wave * 256 + lane;

  // Kick off the DMA for column block 0 into buffer 0.
  {
#pragma unroll
    for (int j = 0; j < 8; ++j) {
      int c16 = cbase16 + j * 32;
      int n = c16 >> 4, c4 = (c16 & 15) << 2;
      unsigned voff = (unsigned)(((n) * Cc + c4) * 4);
      unsigned ldsa = (unsigned)(uintptr_t)(sB0 + n * SB_STRIDE + c4);
      asm volatile("global_load_async_to_lds_b128 %0, %1, %2"
                   :: "v"(ldsa), "v"(voff), "s"(xtb) : "memory");
    }
  }

  float vals[Kk]; int ids[Kk];
#pragma unroll
  for (int i = 0; i < Kk; ++i) { vals[i] = -INFINITY; ids[i] = 0; }

  for (int cb = 0; cb < Nn / 64; ++cb) {
    // Wait for this wave's DMA into the current buffer, then sync the WG so
    // every wave's deposits (and last iteration's top-K reads) are complete.
    asm volatile("s_wait_asynccnt 0" ::: "memory");
    __syncthreads();

    float* cur = (cb & 1) ? sB1 : sB0;
    // Prefetch next column block into the other buffer (overlaps with WMMA).
    if (cb + 1 < Nn / 64) {
      float* nxt = (cb & 1) ? sB0 : sB1;
#pragma unroll
      for (int j = 0; j < 8; ++j) {
        int c16 = cbase16 + j * 32;
        int n = c16 >> 4, c4 = (c16 & 15) << 2;
        unsigned voff = (unsigned)((((cb + 1) * 64 + n) * Cc + c4) * 4);
        unsigned ldsa = (unsigned)(uintptr_t)(nxt + n * SB_STRIDE + c4);
        asm volatile("global_load_async_to_lds_b128 %0, %1, %2"
                     :: "v"(ldsa), "v"(voff), "s"(xtb) : "memory");
      }
    }

    // Each wave owns one 16-wide column tile, both 16-row tiles.
    v8f acc0 = {0.f,0.f,0.f,0.f,0.f,0.f,0.f,0.f};
    v8f acc1 = {0.f,0.f,0.f,0.f,0.f,0.f,0.f,0.f};
#pragma unroll
    for (int ks = 0; ks < 16; ++ks) {
      int c = ks * 4 + lh * 2;                       // K pair for this lane half
      v2f a0 = *(const v2f*)(sA + lm * SA_STRIDE + c);
      v2f a1 = *(const v2f*)(sA + (lm + 16) * SA_STRIDE + c);
      v2f bb = *(const v2f*)(cur + (wave * 16 + lm) * SB_STRIDE + c);
      acc0 = wmma4(a0, bb, acc0);
      acc1 = wmma4(a1, bb, acc1);
    }
    // neg_dist = 2*dot - ||a||^2 - ||b||^2 ; stage into sD.
    float xb = xx[b * Nn + cb * 64 + wave * 16 + lm];   // L2-resident
#pragma unroll
    for (int r = 0; r < 8; ++r) {
      int m0 = r + 8 * lh;
      sD[m0 * SD_STRIDE + wave * 16 + lm]        = 2.f * acc0[r] - sxxA[m0]      - xb;
      sD[(m0 + 16) * SD_STRIDE + wave * 16 + lm] = 2.f * acc1[r] - sxxA[m0 + 16] - xb;
    }
    __syncthreads();

    // Top-K scan: thread -> (row = t%32, 16-col segment = t/32).
    {
      const int mr = t & 31, seg = t >> 5;
      const int cbaseN = cb * 64 + seg * 16;
      for (int j = 0; j < 16; ++j) {
        float v = sD[mr * SD_STRIDE + seg * 16 + j];
        topk_insert(vals, ids, v, cbaseN + j);
      }
    }
  }
  __syncthreads();   // B buffers are dead now; reuse them as merge buffers

  // Merge the 4 per-segment lists of each row.
  {
    const int mr = t & 31, seg = t >> 5;
#pragma unroll
    for (int j = 0; j < Kk; ++j) {
      mVal[(mr * 4 + seg) * Kk + j] = vals[j];
      mIdx[(mr * 4 + seg) * Kk + j] = ids[j];
    }
  }
  __syncthreads();
  if (t < 32) {                 // t<32 -> seg==0: this thread already holds list 0
    for (int s = 1; s < 4; ++s)
      for (int j = 0; j < Kk; ++j)
        topk_insert(vals, ids, mVal[(t * 4 + s) * Kk + j], mIdx[(t * 4 + s) * Kk + j]);
    int* op = idx_out + ((size_t)(b * Nn + row0 + t)) * Kk;
    for (int j = 0; j < Kk; ++j) op[j] = ids[j];
  }
}

// ---------------------------------------------------------------------------
// K2: gathered edge GEMM (one wave per point): Y(32x64) = G(32x128) * Wcomb.
// Rows 0..19 of G are xt[idx[k]] (cols 0-63) / xt[n] (cols 64-127); rows >=20
// are clamped duplicates and masked out of the epilogue.
// Produces per-(b,o,n) max/min over k, plus deterministic stat partials.
// ---------------------------------------------------------------------------
#define SW_STRIDE 132

__global__ __launch_bounds__(256) void edgeconv_kernel(const float* __restrict__ xt,
                                                       const float* __restrict__ Wt,
                                                       const int* __restrict__ idxp,
                                                       float* __restrict__ ymax,
                                                       float* __restrict__ ymin,
                                                       float* __restrict__ partial) {
  __shared__ float sW[Oo * SW_STRIDE];
  __shared__ float sSum[8 * Oo];
  __shared__ float sSq[8 * Oo];
  const int t = threadIdx.x;

  for (int f = t; f < Oo * 32; f += 256) {
    int o = f >> 5, c4 = (f & 31) << 2;
    const float4 v = *(const float4*)(Wt + o * 128 + c4);
    sW[o * SW_STRIDE + c4 + 0] = v.x;
    sW[o * SW_STRIDE + c4 + 1] = v.y;
    sW[o * SW_STRIDE + c4 + 2] = v.z;
    sW[o * SW_STRIDE + c4 + 3] = v.w;
  }
  __syncthreads();

  const int wave = t >> 5, lane = t & 31, lm = lane & 15, lh = lane >> 4;
  const int p = blockIdx.x * 8 + wave;           // global point id
  const int b = p >> 12, n = p & (Nn - 1);

  const int* ip = idxp + (size_t)p * Kk;
  const int i0 = ip[lm];                          // rows 0..15
  const int i1 = ip[min(lm + 16, Kk - 1)];        // rows 16..31 (clamped pad)
  const float* rowA0 = xt + ((size_t)(b * Nn) + i0) * Cc;
  const float* rowA1 = xt + ((size_t)(b * Nn) + i1) * Cc;
  const float* rowC  = xt + (size_t)p * Cc;

  v8f acc[8];
#pragma unroll
  for (int i = 0; i < 8; ++i) acc[i] = (v8f){0.f,0.f,0.f,0.f,0.f,0.f,0.f,0.f};

#pragma unroll 4
  for (int ks = 0; ks < 32; ++ks) {
    int c = ks * 4 + lh * 2;                      // 0..127, uniform side per ks
    v2f a0, a1;
    if (c < 64) {                                 // neighbor features (gather)
      a0 = *(const v2f*)(rowA0 + c);
      a1 = *(const v2f*)(rowA1 + c);
    } else {                                      // center features
      a0 = *(const v2f*)(rowC + (c - 64));
      a1 = a0;
    }
#pragma unroll
    for (int nt = 0; nt < 4; ++nt) {
      v2f bb = *(const v2f*)(sW + (nt * 16 + lm) * SW_STRIDE + c);
      acc[nt]     = wmma4(a0, bb, acc[nt]);
      acc[4 + nt] = wmma4(a1, bb, acc[4 + nt]);
    }
  }

  // Epilogue: per-o max/min/sum/sumsq over valid k (k<20).
#pragma unroll
  for (int nt = 0; nt < 4; ++nt) {
    float mx = -INFINITY, mn = INFINITY, s1 = 0.f, s2 = 0.f;
    v8f A0 = acc[nt], A1 = acc[4 + nt];
#pragma unroll
    for (int r = 0; r < 8; ++r) {                 // k = r + 8*lh  (0..15, valid)
      float v = A0[r];
      mx = fmaxf(mx, v); mn = fminf(mn, v); s1 += v; s2 += v * v;
    }
    if (lh == 0) {
#pragma unroll
      for (int r = 0; r < 4; ++r) {               // k = 16+r (16..19 valid)
        float v = A1[r];
        mx = fmaxf(mx, v); mn = fminf(mn, v); s1 += v; s2 += v * v;
      }
    }
    mx = fmaxf(mx, __shfl_xor(mx, 16, 32));
    mn = fminf(mn, __shfl_xor(mn, 16, 32));
    s1 += __shfl_xor(s1, 16, 32);
    s2 += __shfl_xor(s2, 16, 32);
    if (lh == 0) {
      int o = nt * 16 + lm;
      size_t a = ((size_t)(b * Oo + o)) * Nn + n;
      ymax[a] = mx;
      ymin[a] = mn;
      sSum[wave * Oo + o] = s1;
      sSq[wave * Oo + o]  = s2;
    }
  }
  __syncthreads();
  if (t < 128) {                                  // deterministic per-WG partials
    int o = t & 63, which = t >> 6;
    const float* src = which ? sSq : sSum;
    float a = 0.f;
    for (int w = 0; w < 8; ++w) a += src[w * Oo + o];
    partial[(size_t)blockIdx.x * 128 + t] = a;
  }
}

// ---------------------------------------------------------------------------
// K3: deterministic fixed-order reduction of stat partials (4096 WGs x 128).
// ---------------------------------------------------------------------------
__global__ __launch_bounds__(256) void reduce_stats_kernel(const float* __restrict__ partial,
                                                           float* __restrict__ gstats) {
  __shared__ float red[256];
  const int t = threadIdx.x;
  const int ch = blockIdx.x;                      // 0..127
  float a = 0.f;
  for (int wg = t; wg < 4096; wg += 256) a += partial[(size_t)wg * 128 + ch];
  red[t] = a;
  __syncthreads();
  for (int s = 128; s > 0; s >>= 1) {
    if (t < s) red[t] += red[t + s];
    __syncthreads();
  }
  if (t == 0) gstats[ch] = red[0];
}

// ---------------------------------------------------------------------------
// K4: BN affine + leaky-ReLU + max-over-k via monotonicity (max if slope>=0,
// min otherwise) -> out (B,O,N).
// ---------------------------------------------------------------------------
__global__ __launch_bounds__(256) void finalize_kernel(const float* __restrict__ ymax,
                                                       const float* __restrict__ ymin,
                                                       const float* __restrict__ gstats,
                                                       const float* __restrict__ gamma,
                                                       const float* __restrict__ beta,
                                                       float* __restrict__ out) {
  const int tid = blockIdx.x * 256 + threadIdx.x;
  const int o = (tid >> 12) & 63;
  const float cnt = (float)(Bb * Nn * Kk);
  float mean = gstats[o] / cnt;
  float var  = gstats[64 + o] / cnt - mean * mean;
  float rstd = rsqrtf(var + 1e-5f);
  float s = gamma[o] * rstd;
  float y0 = (s >= 0.f) ? ymax[tid] : ymin[tid];
  float v = (y0 - mean) * s + beta[o];
  out[tid] = (v >= 0.f) ? v : 0.2f * v;
}

// ---------------------------------------------------------------------------
extern "C" void kernel_launch(void* const* d_in, const int* in_sizes, int n_in,
                              void* d_out, int out_size, void* d_ws, size_t ws_size,
                              hipStream_t stream) {
  const float* x     = (const float*)d_in[0];   // (B,C,N)
  const float* W     = (const float*)d_in[1];   // (O,2C)
  const float* gamma = (const float*)d_in[2];   // (O,)
  const float* beta  = (const float*)d_in[3];   // (O,)
  float* out = (float*)d_out;                   // (B,O,N)

  // Workspace carve-up (all float-sized slots; ~30.5 MB total).
  float* xt      = (float*)d_ws;                 // B*N*C    = 2,097,152
  float* xx      = xt + (size_t)Bb * Nn * Cc;    // B*N      =    32,768
  float* Wt      = xx + (size_t)Bb * Nn;         // 64*128   =     8,192
  float* gstats  = Wt + 64 * 128;                // 128
  float* partial = gstats + 128;                 // 4096*128 =   524,288
  int*   idx     = (int*)(partial + 4096 * 128); // B*N*K    =   655,360
  float* ymax    = (float*)(idx + (size_t)Bb * Nn * Kk);   // 2,097,152
  float* ymin    = ymax + (size_t)Bb * Oo * Nn;            // 2,097,152

  prep_x_kernel<<<(Bb * Nn) / 256, 256, 0, stream>>>(x, xt, xx);
  prep_w_kernel<<<1, 256, 0, stream>>>(W, Wt, gstats);

  dim3 g1(Nn / 32, Bb);
  dist_topk_kernel<<<g1, 128, 0, stream>>>(xt, xx, idx);

  edgeconv_kernel<<<(Bb * Nn) / 8, 256, 0, stream>>>(xt, Wt, idx, ymax, ymin, partial);
  reduce_stats_kernel<<<128, 256, 0, stream>>>(partial, gstats);
  finalize_kernel<<<(Bb * Oo * Nn) / 256, 256, 0, stream>>>(ymax, ymin, gstats,
                                                            gamma, beta, out);
}